// GMPooling_17695265259978
// MI455X (gfx1250) — compile-verified
//
#include <hip/hip_runtime.h>
#include <hip/hip_bf16.h>

typedef __attribute__((ext_vector_type(2))) float v2f;
typedef __attribute__((ext_vector_type(8))) float v8f;

// ---------------------------------------------------------------------------
// K1: row L2-normalize. 16 threads per row (4 float4 lanes), half-wave shfl
// reduction, fully coalesced 256B per row.
// ---------------------------------------------------------------------------
__global__ __launch_bounds__(256) void normalize_kernel(
    const float* __restrict__ x, float* __restrict__ xn, int N) {
  const int t = threadIdx.x;
  const int sub = t & 15;        // float4 slot within row
  const int rowInBlk = t >> 4;   // 16 rows per block
  const long row = (long)blockIdx.x * 16 + rowInBlk;
  if (row >= N) return;
  const float4 v = ((const float4*)(x + row * 64))[sub];
  float ss = v.x * v.x + v.y * v.y + v.z * v.z + v.w * v.w;
#pragma unroll
  for (int m = 1; m <= 8; m <<= 1) ss += __shfl_xor(ss, m, 32);
  const float inv = 1.0f / fmaxf(sqrtf(ss), 1e-12f);
  float4 o;
  o.x = v.x * inv; o.y = v.y * inv; o.z = v.z * inv; o.w = v.w * inv;
  ((float4*)(xn + row * 64))[sub] = o;
}

// ---------------------------------------------------------------------------
// K2: edge scoring with FP32 WMMA. One wave = 16 edges.
//   A (16x4 f32): lane m (0..15) holds src_row[m][4k+0,1], lane 16+m holds
//   src_row[m][4k+2,3]  (ISA 7.12.2 32-bit A layout). B mirrors with dst rows
//   so D = A x B^T accumulates all 16x16 dots; we need the diagonal.
//   16 chained V_WMMA_F32_16X16X4_F32 cover K=64 at full f32 precision.
// ---------------------------------------------------------------------------
__global__ __launch_bounds__(256) void edge_score_kernel(
    const float* __restrict__ xn, const int* __restrict__ src,
    const int* __restrict__ dst, const float* __restrict__ score_cut,
    int* __restrict__ sOut, int* __restrict__ tOut, int E) {
  __shared__ float diag[8][32][8];  // 8 waves/block * 32 lanes * 8 acc = 8KB
  const int tid = threadIdx.x;
  const int waveInBlk = tid >> 5;
  const int lane = tid & 31;
  const long wave = (long)blockIdx.x * 8 + waveInBlk;
  const int m = lane & 15;
  const int koff = (lane >> 4) << 1;  // 0 for lanes 0-15, 2 for 16-31
  long e = wave * 16 + m;
  const long ec = (e < E) ? e : (long)E - 1;
  const int s = src[ec];
  const int d = dst[ec];
  const float* __restrict__ arow = xn + (long)s * 64;
  const float* __restrict__ brow = xn + (long)d * 64;

  v8f c = {0.f, 0.f, 0.f, 0.f, 0.f, 0.f, 0.f, 0.f};
#pragma unroll
  for (int k = 0; k < 16; ++k) {
    v2f a, b;
    a.x = arow[4 * k + koff];
    a.y = arow[4 * k + koff + 1];
    b.x = brow[4 * k + koff];
    b.y = brow[4 * k + koff + 1];
    // 8 args: (neg_a, A, neg_b, B, c_mod, C, reuse_a, reuse_b)
    c = __builtin_amdgcn_wmma_f32_16x16x4_f32(false, a, false, b, (short)0, c,
                                              false, false);
  }
#pragma unroll
  for (int j = 0; j < 8; ++j) diag[waveInBlk][lane][j] = c[j];
  __syncthreads();

  if (lane < 16 && e < E) {
    // Diagonal D[M=lane][N=lane]: M<8 -> VGPR M @ lane M; M>=8 -> VGPR M-8 @ lane M+16
    const float dot = (lane < 8) ? diag[waveInBlk][lane][lane]
                                 : diag[waveInBlk][lane + 16][lane - 8];
    float y = 0.999999f * dot;
    y = fminf(fmaxf(y, -0.99999988f), 0.99999988f);
    const float lik = 0.5f * __logf((1.0f + y) / (1.0f - y));  // arctanh
    const bool keep = (lik >= score_cut[0]) && (s != d);
    sOut[e] = keep ? s : 0;   // masked edges become (0,0) self-loops
    tOut[e] = keep ? d : 0;
  }
}

// ---------------------------------------------------------------------------
// Connected components: min-label propagation (fixed iteration count so the
// whole pipeline is graph-capturable and deterministic in work).
// ---------------------------------------------------------------------------
__global__ __launch_bounds__(256) void cc_init_kernel(int* __restrict__ labels, int N) {
  const long i = (long)blockIdx.x * 256 + threadIdx.x;
  if (i < N) labels[i] = (int)i;
}

__global__ __launch_bounds__(256) void cc_relax_kernel(
    const int* __restrict__ s, const int* __restrict__ t,
    int* __restrict__ labels, int E) {
  const long e = (long)blockIdx.x * 256 + threadIdx.x;
  if (e >= E) return;
  const int a = s[e], b = t[e];
  const int la = labels[a], lb = labels[b];
  if (la < lb)      atomicMin(&labels[b], la);
  else if (lb < la) atomicMin(&labels[a], lb);
}

__global__ __launch_bounds__(256) void cc_jump_kernel(int* __restrict__ labels, int N) {
  const long i = (long)blockIdx.x * 256 + threadIdx.x;
  if (i >= N) return;
  const int l = labels[i];
  const int ll = labels[l];
  if (ll < l) atomicMin(&labels[i], ll);
}

// ---------------------------------------------------------------------------
// Component sizes, per-batch has-valid, final label assignment.
// ---------------------------------------------------------------------------
__global__ __launch_bounds__(256) void comp_size_kernel(
    const int* __restrict__ labels, int* __restrict__ comp_size, int N) {
  const long i = (long)blockIdx.x * 256 + threadIdx.x;
  if (i < N) atomicAdd(&comp_size[labels[i]], 1);
}

__global__ __launch_bounds__(256) void has_valid_kernel(
    const int* __restrict__ labels, const int* __restrict__ comp_size,
    const int* __restrict__ batch, const int* __restrict__ min_size,
    int* __restrict__ hv, int N) {
  const long i = (long)blockIdx.x * 256 + threadIdx.x;
  if (i >= N) return;
  if (comp_size[labels[i]] >= min_size[0]) atomicOr(&hv[batch[i]], 1);
}

__global__ __launch_bounds__(256) void final_label_kernel(
    const int* __restrict__ labels, const int* __restrict__ comp_size,
    const int* __restrict__ batch, const int* __restrict__ min_size,
    const int* __restrict__ hv, int* __restrict__ flabels,
    float* __restrict__ out_labels, int N) {
  const long i = (long)blockIdx.x * 256 + threadIdx.x;
  if (i >= N) return;
  const int l = labels[i];
  const bool noise = comp_size[l] < min_size[0];
  const int b = batch[i];
  const bool whole = (hv[b] == 0);
  const int lab2 = whole ? (N + b) : (noise ? (2 * N) : l);
  flabels[i] = lab2;
  out_labels[i] = (float)lab2;
}

// ---------------------------------------------------------------------------
// Segment reductions for centroids / cbatch.
// ---------------------------------------------------------------------------
__global__ __launch_bounds__(256) void accum_kernel(
    const float* __restrict__ xn, const int* __restrict__ flabels,
    const int* __restrict__ batch, float* __restrict__ counts,
    float* __restrict__ sums, int* __restrict__ cbatch, int N) {
  const long gid = (long)blockIdx.x * 256 + threadIdx.x;
  const long node = gid >> 6;
  const int j = (int)(gid & 63);
  if (node >= N) return;
  const int l = flabels[node];
  atomicAdd(&sums[(long)l * 64 + j], xn[node * 64 + j]);
  if (j == 0) {
    atomicAdd(&counts[l], 1.0f);
    atomicMax(&cbatch[l], batch[node]);
  }
}

__global__ __launch_bounds__(256) void centroid_kernel(
    const float* __restrict__ counts, const float* __restrict__ sums,
    float* __restrict__ out, long sent) {
  const long gid = (long)blockIdx.x * 256 + threadIdx.x;
  if (gid >= sent * 64) return;
  const long seg = gid >> 6;
  out[gid] = sums[gid] / fmaxf(counts[seg], 1.0f);
}

__global__ __launch_bounds__(256) void meta_kernel(
    const float* __restrict__ counts, const int* __restrict__ cbatch,
    float* __restrict__ out_cbatch, float* __restrict__ out_valid, long sent) {
  const long seg = (long)blockIdx.x * 256 + threadIdx.x;
  if (seg >= sent) return;
  const bool valid = counts[seg] > 0.0f;
  out_cbatch[seg] = valid ? (float)cbatch[seg] : -1.0f;
  out_valid[seg] = valid ? 1.0f : 0.0f;
}

// ---------------------------------------------------------------------------
extern "C" void kernel_launch(void* const* d_in, const int* in_sizes, int n_in,
                              void* d_out, int out_size, void* d_ws, size_t ws_size,
                              hipStream_t stream) {
  const float* x = (const float*)d_in[0];
  const int* edge_index = (const int*)d_in[1];
  const int* batch = (const int*)d_in[2];
  const float* score_cut = (const float*)d_in[3];
  const int* min_size = (const int*)d_in[4];

  const int N = in_sizes[2];
  const int E = in_sizes[1] / 2;
  const long sent = 2L * N;
  const long nseg = sent + 1;

  const int* src = edge_index;
  const int* dst = edge_index + E;

  // Workspace carve-up (256B aligned).
  char* ws = (char*)d_ws;
  size_t off = 0;
  auto carve = [&](size_t bytes) -> char* {
    char* p = ws + off;
    off = (off + bytes + 255) & ~(size_t)255;
    return p;
  };
  float* xn        = (float*)carve((size_t)N * 64 * sizeof(float));
  int*   sArr      = (int*)carve((size_t)E * sizeof(int));
  int*   tArr      = (int*)carve((size_t)E * sizeof(int));
  int*   labels    = (int*)carve((size_t)N * sizeof(int));
  int*   comp_size = (int*)carve((size_t)N * sizeof(int));
  int*   hv        = (int*)carve(64 * sizeof(int));
  int*   flabels   = (int*)carve((size_t)N * sizeof(int));
  float* counts    = (float*)carve((size_t)nseg * sizeof(float));
  int*   cbatch    = (int*)carve((size_t)nseg * sizeof(int));
  float* sums      = (float*)carve((size_t)nseg * 64 * sizeof(float));
  (void)ws_size;

  float* out = (float*)d_out;
  float* out_centroids = out;                     // sent*64
  float* out_cbatch    = out + sent * 64;         // sent
  float* out_labels    = out_cbatch + sent;       // N
  float* out_valid     = out_labels + N;          // sent

  // 1) normalize rows
  normalize_kernel<<<(N + 15) / 16, 256, 0, stream>>>(x, xn, N);

  // 2) WMMA edge scoring -> masked edge lists
  {
    const long waves = (E + 15) / 16;
    const int blocks = (int)((waves + 7) / 8);
    edge_score_kernel<<<blocks, 256, 0, stream>>>(xn, src, dst, score_cut,
                                                  sArr, tArr, E);
  }

  // 3) connected components, fixed iteration count (deterministic, capturable)
  cc_init_kernel<<<(N + 255) / 256, 256, 0, stream>>>(labels, N);
  for (int it = 0; it < 40; ++it) {
    cc_relax_kernel<<<(E + 255) / 256, 256, 0, stream>>>(sArr, tArr, labels, E);
    cc_jump_kernel<<<(N + 255) / 256, 256, 0, stream>>>(labels, N);
  }

  // 4) component sizes, has-valid per batch, final labels
  hipMemsetAsync(comp_size, 0, (size_t)N * sizeof(int), stream);
  comp_size_kernel<<<(N + 255) / 256, 256, 0, stream>>>(labels, comp_size, N);
  hipMemsetAsync(hv, 0, 64 * sizeof(int), stream);
  has_valid_kernel<<<(N + 255) / 256, 256, 0, stream>>>(labels, comp_size,
                                                        batch, min_size, hv, N);
  final_label_kernel<<<(N + 255) / 256, 256, 0, stream>>>(
      labels, comp_size, batch, min_size, hv, flabels, out_labels, N);

  // 5) segment reductions -> centroids / cbatch / valid
  hipMemsetAsync(counts, 0, (size_t)nseg * sizeof(float), stream);
  hipMemsetAsync(sums, 0, (size_t)nseg * 64 * sizeof(float), stream);
  hipMemsetAsync(cbatch, 0xFF, (size_t)nseg * sizeof(int), stream);  // -1
  {
    const long work = (long)N * 64;
    accum_kernel<<<(int)((work + 255) / 256), 256, 0, stream>>>(
        xn, flabels, batch, counts, sums, cbatch, N);
  }
  {
    const long work = sent * 64;
    centroid_kernel<<<(int)((work + 255) / 256), 256, 0, stream>>>(
        counts, sums, out_centroids, sent);
  }
  meta_kernel<<<(int)((sent + 255) / 256), 256, 0, stream>>>(
      counts, cbatch, out_cbatch, out_valid, sent);
}